// BaselineGraphSAGE_28346784153656
// MI455X (gfx1250) — compile-verified
//
#include <hip/hip_runtime.h>

// ---------------------------------------------------------------------------
// GraphSAGE (2x SAGEConv) forward for MI455X / gfx1250.
//  - scatter-mean via L2-resident fp32 atomics (accumulators fit in 192MB L2)
//  - fused dual GEMM  [agg*invdeg | x] @ [Wl.T ; Wr.T] + b  via
//    V_WMMA_F32_16X16X4_F32 (exact fp32 matrix math, wave32)
// ---------------------------------------------------------------------------

typedef __attribute__((ext_vector_type(2))) float v2f;
typedef __attribute__((ext_vector_type(8))) float v8f;

#define K_IN 128           // feature width of both layers' inputs
#define AS_STRIDE 260      // LDS row stride (floats); 260 % 64 == 4 -> no bank conflicts for b64 reads

// ---------------- utility kernels ----------------

__global__ void zero_f32(float* __restrict__ p, long n) {
  long i = (long)blockIdx.x * blockDim.x + threadIdx.x;
  long stride = (long)gridDim.x * blockDim.x;
  for (; i < n; i += stride) p[i] = 0.0f;
}

__global__ void edge_deg(const long long* __restrict__ dst, int E,
                         float* __restrict__ deg) {
  int e = blockIdx.x * blockDim.x + threadIdx.x;
  if (e < E) atomicAdd(&deg[(int)dst[e]], 1.0f);
}

__global__ void inv_deg(float* __restrict__ deg, int n) {
  int i = blockIdx.x * blockDim.x + threadIdx.x;
  if (i < n) deg[i] = 1.0f / fmaxf(deg[i], 1.0f);
}

// One wave per edge: gather a 128-float row of `feat` at src, atomically add
// it into the accumulator row at dst.  Accumulator is L2-resident (51 MB).
__global__ void edge_scatter(const long long* __restrict__ src,
                             const long long* __restrict__ dst, int E,
                             const float* __restrict__ feat,
                             float* __restrict__ agg) {
  int e = blockIdx.x * (blockDim.x >> 5) + (threadIdx.x >> 5);
  if (e >= E) return;
  int lane = threadIdx.x & 31;
  int s = (int)src[e];
  int d = (int)dst[e];
  const float4 v = *(const float4*)(feat + (size_t)s * K_IN + lane * 4);
  float* o = agg + (size_t)d * K_IN + lane * 4;
  atomicAdd(o + 0, v.x);
  atomicAdd(o + 1, v.y);
  atomicAdd(o + 2, v.z);
  atomicAdd(o + 3, v.w);
}

// ---------------- fused SAGEConv dense kernel ----------------
// Out[m, :Cout] = relu?( (Aagg[m]*invdeg[m]) @ Wl.T + bl + Aself[m] @ Wr.T )
// Block: 16 rows of output, ceil(Cout/16) waves, one 16x16 tile per wave.
// A rows (16 x 256 fp32) staged in LDS; 64x V_WMMA_F32_16X16X4_F32 per tile.
__global__ void sage_gemm(const float* __restrict__ Aagg,
                          const float* __restrict__ invdeg,
                          const float* __restrict__ Aself,
                          const float* __restrict__ Wl,   // [C, 128] row-major
                          const float* __restrict__ bl,   // [C]
                          const float* __restrict__ Wr,   // [C, 128] row-major
                          float* __restrict__ Out,        // [N, ldOut]
                          int n_nodes, int Cout, int ldOut, int do_relu) {
  __shared__ __align__(16) float As[16 * AS_STRIDE];

  const int m0 = blockIdx.x * 16;

  // Stage A = [ agg*invdeg | self ] : 16 rows x 256 cols of fp32.
  for (int idx = threadIdx.x; idx < 16 * 2 * K_IN; idx += blockDim.x) {
    int r = idx >> 8;        // idx / 256
    int k = idx & 255;       // idx % 256
    int node = m0 + r;
    float v = 0.0f;
    if (node < n_nodes) {
      if (k < K_IN) v = Aagg[(size_t)node * K_IN + k] * invdeg[node];
      else          v = Aself[(size_t)node * K_IN + (k - K_IN)];
    }
    As[r * AS_STRIDE + k] = v;
  }
  __syncthreads();

  const int lane  = threadIdx.x & 31;
  const int wv    = threadIdx.x >> 5;
  const int m     = lane & 15;        // A row within tile / D column index base
  const int khalf = lane >> 4;        // 0: K0..1, 1: K2..3 (f32 16x4 A layout)
  const int col   = wv * 16 + (lane & 15);
  const bool colok = col < Cout;
  const int colc  = colok ? col : 0;

  const float* wlrow = Wl + (size_t)colc * K_IN;  // B[k][n] = W[n][k]
  const float* wrrow = Wr + (size_t)colc * K_IN;
  const float* arow  = &As[m * AS_STRIDE];

  v8f acc = {0.f, 0.f, 0.f, 0.f, 0.f, 0.f, 0.f, 0.f};

  // agg*invdeg  @ Wl.T  (K = 0..127 of staged A)
#pragma unroll
  for (int kk = 0; kk < K_IN; kk += 4) {
    int ka = kk + khalf * 2;
    v2f a = *(const v2f*)(arow + ka);          // ds_load_b64
    v2f b = *(const v2f*)(wlrow + ka);         // global_load_b64 (L2-hot)
    acc = __builtin_amdgcn_wmma_f32_16x16x4_f32(false, a, false, b,
                                                (short)0, acc, false, false);
  }
  // self @ Wr.T  (K = 128..255 of staged A)
#pragma unroll
  for (int kk = 0; kk < K_IN; kk += 4) {
    int ka = kk + khalf * 2;
    v2f a = *(const v2f*)(arow + K_IN + ka);
    v2f b = *(const v2f*)(wrrow + ka);
    acc = __builtin_amdgcn_wmma_f32_16x16x4_f32(false, a, false, b,
                                                (short)0, acc, false, false);
  }

  // Epilogue: bias, relu, masked store.  D layout: lane n = l%16,
  // VGPR v holds row v (lanes 0-15) / row v+8 (lanes 16-31).
  float bias = colok ? bl[col] : 0.0f;
  int rbase = m0 + (khalf ? 8 : 0);
#pragma unroll
  for (int v = 0; v < 8; ++v) {
    int row = rbase + v;
    float val = acc[v] + bias;
    if (do_relu) val = fmaxf(val, 0.0f);
    if (colok && row < n_nodes) Out[(size_t)row * ldOut + col] = val;
  }
}

// ---------------- host-side launch ----------------

extern "C" void kernel_launch(void* const* d_in, const int* in_sizes, int n_in,
                              void* d_out, int out_size, void* d_ws, size_t ws_size,
                              hipStream_t stream) {
  const float*     x    = (const float*)d_in[0];
  const long long* ei   = (const long long*)d_in[1];   // int64 [2, E]
  const float*     Wl1  = (const float*)d_in[2];
  const float*     bl1  = (const float*)d_in[3];
  const float*     Wr1  = (const float*)d_in[4];
  const float*     Wl2  = (const float*)d_in[5];
  const float*     bl2  = (const float*)d_in[6];
  const float*     Wr2  = (const float*)d_in[7];
  float*           out  = (float*)d_out;

  const int N = in_sizes[0] / K_IN;
  const int E = in_sizes[1] / 2;
  const long long* src = ei;
  const long long* dst = ei + E;

  // Workspace layout (floats): [deg | agg (N*128) | h (N*128)]
  float* ws = (float*)d_ws;
  const size_t DEG_OFF = 0;
  const size_t AGG_OFF = ((size_t)N + 511) & ~(size_t)511;   // 16B-aligned pad
  const size_t H_OFF   = AGG_OFF + (size_t)N * K_IN;
  float* deg = ws + DEG_OFF;
  float* agg = ws + AGG_OFF;
  float* h   = ws + H_OFF;

  // ---- degree + layer-1 aggregation ----
  zero_f32<<<1024, 256, 0, stream>>>(deg, (long)N);
  zero_f32<<<4096, 256, 0, stream>>>(agg, (long)N * K_IN);
  edge_deg<<<(E + 255) / 256, 256, 0, stream>>>(dst, E, deg);
  edge_scatter<<<(E + 7) / 8, 256, 0, stream>>>(src, dst, E, x, agg);
  inv_deg<<<(N + 255) / 256, 256, 0, stream>>>(deg, N);

  // ---- layer 1 dense: h = relu(mean_agg @ Wl1.T + bl1 + x @ Wr1.T) ----
  sage_gemm<<<(N + 15) / 16, 256, 0, stream>>>(agg, deg, x, Wl1, bl1, Wr1,
                                               h, N, 128, 128, 1);

  // ---- layer-2 aggregation over h ----
  zero_f32<<<4096, 256, 0, stream>>>(agg, (long)N * K_IN);
  edge_scatter<<<(E + 7) / 8, 256, 0, stream>>>(src, dst, E, h, agg);

  // ---- layer 2 dense: out = mean_agg @ Wl2.T + bl2 + h @ Wr2.T  (40 cols) ----
  sage_gemm<<<(N + 15) / 16, 96, 0, stream>>>(agg, deg, h, Wl2, bl2, Wr2,
                                              out, N, 40, 40, 0);
}